// voxel_3d_generator_22522808500262
// MI455X (gfx1250) — compile-verified
//
#include <hip/hip_runtime.h>

// Voxel 3D generator for MI455X (gfx1250, wave32, WMMA).
// Strategy: fold each BatchNorm (affine per column) into an adjacent GEMM's
// weights once its global stats are known; recompute the cheap 13-feature
// front-end per pass instead of spilling N x 64 intermediates to HBM.
// All GEMMs run on V_WMMA_F32_16X16X4_F32 (native f32 matrix path).

#define BN_EPS 1e-5f
#define WPB 4            // waves per block in WMMA kernels (128 threads)
#define GBLK 2048        // blocks for WMMA kernels

typedef __attribute__((ext_vector_type(2))) float v2f;
typedef __attribute__((ext_vector_type(8))) float v8f;

__device__ __forceinline__ v8f wmma_f32_16x16x4(v2f a, v2f b, v8f c) {
  // D = A(16x4) * B(4x16) + C(16x16), all f32, one wave.
  return __builtin_amdgcn_wmma_f32_16x16x4_f32(
      /*neg_a=*/false, a, /*neg_b=*/false, b,
      /*c_mod=*/(short)0, c, /*reuse_a=*/false, /*reuse_b=*/false);
}

// ---------------------------------------------------------------------------
// 13-feature front end: [points(4), xyz - voxel_mean(3), xyz - center(3),
//                        normal(3)] padded to 16.
// ---------------------------------------------------------------------------
__device__ __forceinline__ void compute_features(
    int i, const float* __restrict__ pts, const float* __restrict__ nrm,
    const int* __restrict__ gid, const int* __restrict__ cinv,
    const float* __restrict__ vox, float f[16]) {
  const float IVXY = 102.4f / 512.0f;   // (51.2 - -51.2)/512
  const float IVZ  = 6.4f / 32.0f;      // (2.4 - -4.0)/32
  float x = pts[i * 4 + 0], y = pts[i * 4 + 1];
  float z = pts[i * 4 + 2], w = pts[i * 4 + 3];
  int v = cinv[i];
  float c  = fmaxf(vox[v * 4 + 3], 1.0f);
  float mx = vox[v * 4 + 0] / c;
  float my = vox[v * 4 + 1] / c;
  float mz = vox[v * 4 + 2] / c;
  float cx = (float)gid[i * 3 + 0] * IVXY - 51.2f;
  float cy = (float)gid[i * 3 + 1] * IVXY - 51.2f;
  float cz = (float)gid[i * 3 + 2] * IVZ  - 4.0f;
  f[0] = x; f[1] = y; f[2] = z; f[3] = w;
  f[4] = x - mx; f[5] = y - my; f[6] = z - mz;
  f[7] = x - cx; f[8] = y - cy; f[9] = z - cz;
  f[10] = nrm[i * 3 + 0]; f[11] = nrm[i * 3 + 1]; f[12] = nrm[i * 3 + 2];
  f[13] = 0.f; f[14] = 0.f; f[15] = 0.f;
}

// Lanes 0..15 each build one 16-wide feature row into the wave's LDS tile
// (stride 20 floats -> conflict-free column reads afterwards).
__device__ __forceinline__ void fill_feature_tile(
    int base, int N, int lane, float* __restrict__ fl,
    const float* __restrict__ pts, const float* __restrict__ nrm,
    const int* __restrict__ gid, const int* __restrict__ cinv,
    const float* __restrict__ vox) {
  if (lane < 16) {
    float f[16];
#pragma unroll
    for (int k = 0; k < 16; k++) f[k] = 0.f;
    int i = base + lane;
    if (i < N) compute_features(i, pts, nrm, gid, cinv, vox, f);
#pragma unroll
    for (int k = 0; k < 16; k++) fl[lane * 20 + k] = f[k];
  }
}

// One wave: 16-row x 64-col GEMM tile via f32 WMMA 16x16x4.
// aT: LDS [16][SA] row-major activations; wB: LDS [K][64]; bias: LDS [64].
template <int KSTEPS, int SA>
__device__ __forceinline__ void gemm_tile(const float* __restrict__ aT,
                                          const float* __restrict__ wB,
                                          const float* __restrict__ bias,
                                          int lane, v8f acc[4]) {
  int col  = lane & 15;
  int koff = (lane >> 4) << 1;  // 0 for lanes 0-15, 2 for lanes 16-31
#pragma unroll
  for (int nt = 0; nt < 4; nt++) {
    float bv = bias[nt * 16 + col];
    v8f c = {bv, bv, bv, bv, bv, bv, bv, bv};
    acc[nt] = c;
  }
#pragma unroll
  for (int ks = 0; ks < KSTEPS; ks++) {
    int kk = ks * 4;
    v2f a;
    a.x = aT[col * SA + kk + koff];      // A[M=col][K=kk+koff]
    a.y = aT[col * SA + kk + koff + 1];
#pragma unroll
    for (int nt = 0; nt < 4; nt++) {
      v2f b;
      b.x = wB[(kk + koff) * 64 + nt * 16 + col];      // B[K][N]
      b.y = wB[(kk + koff + 1) * 64 + nt * 16 + col];
      acc[nt] = wmma_f32_16x16x4(a, b, acc[nt]);
    }
  }
}

// Write C-layout accumulators back to an LDS [16][SA] activation tile w/ relu.
__device__ __forceinline__ void store_relu(float* __restrict__ dst, int SA,
                                           const v8f acc[4], int lane) {
  int col = lane & 15;
  int rb  = (lane >> 4) * 8;
#pragma unroll
  for (int nt = 0; nt < 4; nt++) {
#pragma unroll
    for (int r = 0; r < 8; r++) {
      dst[(rb + r) * SA + nt * 16 + col] = fmaxf(acc[nt][r], 0.f);
    }
  }
}

// ---------------------------------------------------------------------------
// K1: voxel sums + counts (scatter for pc_mean)
// ---------------------------------------------------------------------------
__global__ void k_voxsum(const float* __restrict__ pts,
                         const int* __restrict__ cinv,
                         float* __restrict__ vox, int N) {
  int i = blockIdx.x * blockDim.x + threadIdx.x;
  if (i < N) {
    int v = cinv[i];
    atomicAdd(&vox[v * 4 + 0], pts[i * 4 + 0]);
    atomicAdd(&vox[v * 4 + 1], pts[i * 4 + 1]);
    atomicAdd(&vox[v * 4 + 2], pts[i * 4 + 2]);
    atomicAdd(&vox[v * 4 + 3], 1.0f);
  }
}

// ---------------------------------------------------------------------------
// K2: per-column sum / sumsq of the 13 features (BN0 stats)
// ---------------------------------------------------------------------------
__global__ void k_feat_stats(const float* __restrict__ pts,
                             const float* __restrict__ nrm,
                             const int* __restrict__ gid,
                             const int* __restrict__ cinv,
                             const float* __restrict__ vox,
                             float* __restrict__ s13, int N) {
  int tid = blockIdx.x * blockDim.x + threadIdx.x;
  int stride = gridDim.x * blockDim.x;
  float s[13], q[13];
#pragma unroll
  for (int k = 0; k < 13; k++) { s[k] = 0.f; q[k] = 0.f; }
  for (int i = tid; i < N; i += stride) {
    float f[16];
    compute_features(i, pts, nrm, gid, cinv, vox, f);
#pragma unroll
    for (int k = 0; k < 13; k++) { s[k] += f[k]; q[k] += f[k] * f[k]; }
  }
#pragma unroll
  for (int m = 16; m >= 1; m >>= 1) {
#pragma unroll
    for (int k = 0; k < 13; k++) {
      s[k] += __shfl_xor(s[k], m, 32);
      q[k] += __shfl_xor(q[k], m, 32);
    }
  }
  if ((threadIdx.x & 31) == 0) {
#pragma unroll
    for (int k = 0; k < 13; k++) {
      atomicAdd(&s13[k], s[k]);
      atomicAdd(&s13[16 + k], q[k]);
    }
  }
}

// ---------------------------------------------------------------------------
// K3: fold BN0 into W1:  W1f = diag(s0) W1 (padded 16x64), b1f = t0@W1 + b1
// ---------------------------------------------------------------------------
__global__ void k_fold0(const float* __restrict__ s13,
                        const float* __restrict__ g0, const float* __restrict__ b0,
                        const float* __restrict__ w1, const float* __restrict__ b1,
                        float* __restrict__ w1f, float* __restrict__ b1f, float fN) {
  __shared__ float s0[13], t0[13];
  int j = threadIdx.x;  // 64 threads, one per output column
  if (j < 13) {
    float m  = s13[j] / fN;
    float vv = s13[16 + j] / fN - m * m;
    float sc = g0[j] * rsqrtf(vv + BN_EPS);
    s0[j] = sc;
    t0[j] = b0[j] - m * sc;
  }
  __syncthreads();
  float bb = b1[j];
#pragma unroll
  for (int k = 0; k < 13; k++) {
    bb += t0[k] * w1[k * 64 + j];
    w1f[k * 64 + j] = s0[k] * w1[k * 64 + j];
  }
#pragma unroll
  for (int k = 13; k < 16; k++) w1f[k * 64 + j] = 0.f;
  b1f[j] = bb;
}

// ---------------------------------------------------------------------------
// K4: WMMA GEMM1 -> per-column sum/sumsq of h1 (BN1 stats)
// ---------------------------------------------------------------------------
__global__ __launch_bounds__(128) void k_h1_stats(
    const float* __restrict__ pts, const float* __restrict__ nrm,
    const int* __restrict__ gid, const int* __restrict__ cinv,
    const float* __restrict__ vox, const float* __restrict__ w1f,
    const float* __restrict__ b1f, float* __restrict__ stat, int N) {
  __shared__ float sW[16 * 64];
  __shared__ float sB[64];
  __shared__ float sF[WPB][16 * 20];
  for (int t = threadIdx.x; t < 16 * 64; t += blockDim.x) sW[t] = w1f[t];
  if (threadIdx.x < 64) sB[threadIdx.x] = b1f[threadIdx.x];
  __syncthreads();
  int wave = threadIdx.x >> 5, lane = threadIdx.x & 31, col = lane & 15;
  int gw = blockIdx.x * WPB + wave, nw = gridDim.x * WPB;
  int nch = (N + 15) >> 4;
  float ps[4] = {0, 0, 0, 0}, pq[4] = {0, 0, 0, 0};
  float* fl = sF[wave];
  for (int ch = gw; ch < nch; ch += nw) {
    int base = ch * 16;
    fill_feature_tile(base, N, lane, fl, pts, nrm, gid, cinv, vox);
    v8f acc[4];
    gemm_tile<4, 20>(fl, sW, sB, lane, acc);
    int rb = (lane >> 4) * 8;
#pragma unroll
    for (int r = 0; r < 8; r++) {
      if (base + rb + r < N) {
#pragma unroll
        for (int nt = 0; nt < 4; nt++) {
          float v = acc[nt][r];
          ps[nt] += v;
          pq[nt] += v * v;
        }
      }
    }
  }
#pragma unroll
  for (int nt = 0; nt < 4; nt++) {
    ps[nt] += __shfl_xor(ps[nt], 16, 32);
    pq[nt] += __shfl_xor(pq[nt], 16, 32);
  }
  if (lane < 16) {
#pragma unroll
    for (int nt = 0; nt < 4; nt++) {
      atomicAdd(&stat[nt * 16 + col], ps[nt]);
      atomicAdd(&stat[64 + nt * 16 + col], pq[nt]);
    }
  }
}

// ---------------------------------------------------------------------------
// K5: fold BN1 into (W1f,b1f):  W1g = W1f * s1_col, b1g = b1f*s1 + t1
// ---------------------------------------------------------------------------
__global__ void k_fold1(const float* __restrict__ st,
                        const float* __restrict__ g1, const float* __restrict__ b1,
                        const float* __restrict__ w1f, const float* __restrict__ b1f,
                        float* __restrict__ w1g, float* __restrict__ b1g, float fN) {
  int j = threadIdx.x;  // 64
  float m  = st[j] / fN;
  float vv = st[64 + j] / fN - m * m;
  float s  = g1[j] * rsqrtf(vv + BN_EPS);
  float t  = b1[j] - m * s;
#pragma unroll
  for (int k = 0; k < 16; k++) w1g[k * 64 + j] = w1f[k * 64 + j] * s;
  b1g[j] = b1f[j] * s + t;
}

// ---------------------------------------------------------------------------
// K6: WMMA G1 -> relu -> WMMA G2 (raw W2) -> sum/sumsq of h2pre (BN2 stats)
// ---------------------------------------------------------------------------
__global__ __launch_bounds__(128) void k_h2_stats(
    const float* __restrict__ pts, const float* __restrict__ nrm,
    const int* __restrict__ gid, const int* __restrict__ cinv,
    const float* __restrict__ vox, const float* __restrict__ w1g,
    const float* __restrict__ b1g, const float* __restrict__ w2,
    const float* __restrict__ b2, float* __restrict__ stat, int N) {
  __shared__ float sW1[16 * 64];
  __shared__ float sB1[64];
  __shared__ float sW2[64 * 64];
  __shared__ float sB2[64];
  __shared__ float sF[WPB][16 * 20];
  __shared__ float sA[WPB][16 * 68];
  for (int t = threadIdx.x; t < 16 * 64; t += blockDim.x) sW1[t] = w1g[t];
  for (int t = threadIdx.x; t < 64 * 64; t += blockDim.x) sW2[t] = w2[t];
  if (threadIdx.x < 64) { sB1[threadIdx.x] = b1g[threadIdx.x]; sB2[threadIdx.x] = b2[threadIdx.x]; }
  __syncthreads();
  int wave = threadIdx.x >> 5, lane = threadIdx.x & 31, col = lane & 15;
  int gw = blockIdx.x * WPB + wave, nw = gridDim.x * WPB;
  int nch = (N + 15) >> 4;
  float ps[4] = {0, 0, 0, 0}, pq[4] = {0, 0, 0, 0};
  float* fl = sF[wave];
  float* al = sA[wave];
  for (int ch = gw; ch < nch; ch += nw) {
    int base = ch * 16;
    fill_feature_tile(base, N, lane, fl, pts, nrm, gid, cinv, vox);
    v8f acc[4];
    gemm_tile<4, 20>(fl, sW1, sB1, lane, acc);
    store_relu(al, 68, acc, lane);          // a = relu(bn1 folded)
    gemm_tile<16, 68>(al, sW2, sB2, lane, acc);  // h2pre
    int rb = (lane >> 4) * 8;
#pragma unroll
    for (int r = 0; r < 8; r++) {
      if (base + rb + r < N) {
#pragma unroll
        for (int nt = 0; nt < 4; nt++) {
          float v = acc[nt][r];
          ps[nt] += v;
          pq[nt] += v * v;
        }
      }
    }
  }
#pragma unroll
  for (int nt = 0; nt < 4; nt++) {
    ps[nt] += __shfl_xor(ps[nt], 16, 32);
    pq[nt] += __shfl_xor(pq[nt], 16, 32);
  }
  if (lane < 16) {
#pragma unroll
    for (int nt = 0; nt < 4; nt++) {
      atomicAdd(&stat[nt * 16 + col], ps[nt]);
      atomicAdd(&stat[64 + nt * 16 + col], pq[nt]);
    }
  }
}

// ---------------------------------------------------------------------------
// K7: fold BN2 into W2:  W2g = W2 * s2_col, b2g = b2*s2 + t2
// ---------------------------------------------------------------------------
__global__ void k_fold2(const float* __restrict__ st,
                        const float* __restrict__ g2, const float* __restrict__ b2n,
                        const float* __restrict__ w2, const float* __restrict__ b2,
                        float* __restrict__ w2g, float* __restrict__ b2g, float fN) {
  int j = threadIdx.x;  // 64
  float m  = st[j] / fN;
  float vv = st[64 + j] / fN - m * m;
  float s  = g2[j] * rsqrtf(vv + BN_EPS);
  float t  = b2n[j] - m * s;
  for (int k = 0; k < 64; k++) w2g[k * 64 + j] = w2[k * 64 + j] * s;
  b2g[j] = b2[j] * s + t;
}

// ---------------------------------------------------------------------------
// K8: full forward (G1 -> relu -> G2 -> relu -> G3) + scatter-sum into d_out
// ---------------------------------------------------------------------------
__global__ __launch_bounds__(128) void k_final(
    const float* __restrict__ pts, const float* __restrict__ nrm,
    const int* __restrict__ gid, const int* __restrict__ cinv,
    const float* __restrict__ vox, const float* __restrict__ w1g,
    const float* __restrict__ b1g, const float* __restrict__ w2g,
    const float* __restrict__ b2g, const float* __restrict__ w3,
    const float* __restrict__ b3, float* __restrict__ out, int N) {
  __shared__ float sW1[16 * 64];
  __shared__ float sB1[64];
  __shared__ float sW2[64 * 64];
  __shared__ float sB2[64];
  __shared__ float sW3[64 * 64];
  __shared__ float sB3[64];
  __shared__ float sF[WPB][16 * 20];
  __shared__ float sA[WPB][16 * 68];
  for (int t = threadIdx.x; t < 16 * 64; t += blockDim.x) sW1[t] = w1g[t];
  for (int t = threadIdx.x; t < 64 * 64; t += blockDim.x) { sW2[t] = w2g[t]; sW3[t] = w3[t]; }
  if (threadIdx.x < 64) {
    sB1[threadIdx.x] = b1g[threadIdx.x];
    sB2[threadIdx.x] = b2g[threadIdx.x];
    sB3[threadIdx.x] = b3[threadIdx.x];
  }
  __syncthreads();
  int wave = threadIdx.x >> 5, lane = threadIdx.x & 31, col = lane & 15;
  int gw = blockIdx.x * WPB + wave, nw = gridDim.x * WPB;
  int nch = (N + 15) >> 4;
  float* fl = sF[wave];
  float* al = sA[wave];
  for (int ch = gw; ch < nch; ch += nw) {
    int base = ch * 16;
    fill_feature_tile(base, N, lane, fl, pts, nrm, gid, cinv, vox);
    v8f acc[4];
    gemm_tile<4, 20>(fl, sW1, sB1, lane, acc);
    store_relu(al, 68, acc, lane);               // a  = relu(G1)
    gemm_tile<16, 68>(al, sW2, sB2, lane, acc);
    store_relu(al, 68, acc, lane);               // h2 = relu(G2) (in-place ok)
    gemm_tile<16, 68>(al, sW3, sB3, lane, acc);  // h3
    int rb = (lane >> 4) * 8;
#pragma unroll
    for (int r = 0; r < 8; r++) {
      int i = base + rb + r;
      if (i < N) {
        int v = cinv[i];
#pragma unroll
        for (int nt = 0; nt < 4; nt++) {
          atomicAdd(&out[(size_t)v * 64 + nt * 16 + col], acc[nt][r]);
        }
      }
    }
  }
}

// ---------------------------------------------------------------------------
// K9: divide scatter sums by per-voxel counts
// ---------------------------------------------------------------------------
__global__ void k_div(float* __restrict__ out, const float* __restrict__ vox, int V) {
  int t = blockIdx.x * blockDim.x + threadIdx.x;
  if (t < V * 64) {
    int v = t >> 6;
    out[t] /= fmaxf(vox[v * 4 + 3], 1.0f);
  }
}

// ---------------------------------------------------------------------------
extern "C" void kernel_launch(void* const* d_in, const int* in_sizes, int n_in,
                              void* d_out, int out_size, void* d_ws, size_t ws_size,
                              hipStream_t stream) {
  const float* pts   = (const float*)d_in[0];
  const float* nrm   = (const float*)d_in[1];
  const int*   gid   = (const int*)d_in[2];
  const int*   cinv  = (const int*)d_in[3];
  const float* bn0g  = (const float*)d_in[4];
  const float* bn0b  = (const float*)d_in[5];
  const float* w1    = (const float*)d_in[6];
  const float* bias1 = (const float*)d_in[7];
  const float* bn1g  = (const float*)d_in[8];
  const float* bn1b  = (const float*)d_in[9];
  const float* w2    = (const float*)d_in[10];
  const float* bias2 = (const float*)d_in[11];
  const float* bn2g  = (const float*)d_in[12];
  const float* bn2b  = (const float*)d_in[13];
  const float* w3    = (const float*)d_in[14];
  const float* bias3 = (const float*)d_in[15];
  (void)n_in; (void)ws_size;

  int N = in_sizes[0] / 4;
  int V = out_size / 64;
  float* ws  = (float*)d_ws;
  float* out = (float*)d_out;

  // workspace layout (floats)
  size_t o_vox = 0;                       // V*4 : sumx,sumy,sumz,count
  size_t o_s13 = (size_t)V * 4;           // 32  : feature sum[16], sumsq[16]
  size_t o_s1  = o_s13 + 32;              // 128 : h1 sum[64], sumsq[64]
  size_t o_s2  = o_s1 + 128;              // 128 : h2pre sum[64], sumsq[64]
  size_t o_w1f = o_s2 + 128;              // 1024: BN0-folded W1 (16x64)
  size_t o_b1f = o_w1f + 1024;            // 64
  size_t o_w1g = o_b1f + 64;              // 1024: BN0+BN1-folded W1
  size_t o_b1g = o_w1g + 1024;            // 64
  size_t o_w2g = o_b1g + 64;              // 4096: BN2-folded W2
  size_t o_b2g = o_w2g + 4096;            // 64
  size_t used  = o_b2g + 64;

  (void)hipMemsetAsync(d_ws, 0, used * sizeof(float), stream);
  (void)hipMemsetAsync(d_out, 0, (size_t)out_size * sizeof(float), stream);

  float fN = (float)N;

  k_voxsum<<<(N + 255) / 256, 256, 0, stream>>>(pts, cinv, ws + o_vox, N);

  k_feat_stats<<<512, 256, 0, stream>>>(pts, nrm, gid, cinv, ws + o_vox,
                                        ws + o_s13, N);

  k_fold0<<<1, 64, 0, stream>>>(ws + o_s13, bn0g, bn0b, w1, bias1,
                                ws + o_w1f, ws + o_b1f, fN);

  k_h1_stats<<<GBLK, 128, 0, stream>>>(pts, nrm, gid, cinv, ws + o_vox,
                                       ws + o_w1f, ws + o_b1f, ws + o_s1, N);

  k_fold1<<<1, 64, 0, stream>>>(ws + o_s1, bn1g, bn1b, ws + o_w1f, ws + o_b1f,
                                ws + o_w1g, ws + o_b1g, fN);

  k_h2_stats<<<GBLK, 128, 0, stream>>>(pts, nrm, gid, cinv, ws + o_vox,
                                       ws + o_w1g, ws + o_b1g, w2, bias2,
                                       ws + o_s2, N);

  k_fold2<<<1, 64, 0, stream>>>(ws + o_s2, bn2g, bn2b, w2, bias2,
                                ws + o_w2g, ws + o_b2g, fN);

  k_final<<<GBLK, 128, 0, stream>>>(pts, nrm, gid, cinv, ws + o_vox,
                                    ws + o_w1g, ws + o_b1g, ws + o_w2g,
                                    ws + o_b2g, w3, bias3, out, N);

  k_div<<<((size_t)V * 64 + 255) / 256, 256, 0, stream>>>(out, ws + o_vox, V);
}